// ChamferLoss_816043786353
// MI455X (gfx1250) — compile-verified
//
#include <hip/hip_runtime.h>

typedef float v2f __attribute__((ext_vector_type(2)));
typedef float v8f __attribute__((ext_vector_type(8)));

// One wave owns a tile of 16 "query" points (rows) of batch b and scans all
// "ref" points (columns) 16 at a time with V_WMMA_F32_16X16X4_F32.
//   A row m  = [-2qx, -2qy, -2qz, 1]        (held in a v2f fragment per lane)
//   B col n  = [ rx,   ry,   rz, |r|^2]
//   C seed   = |q_m|^2 broadcast across row m
// => D[m][n] = |q|^2 + |r|^2 - 2 q.r  (squared Euclidean distance)
// Row-min is tracked on squared distances (sqrt is monotonic), sqrt applied
// once per row at the end. minout[b][m] = min_n dist(q_m, r_n).
__global__ void chamfer_rowmin_wmma(const float* __restrict__ Q,  // [B, NQ, 3]
                                    const float* __restrict__ R,  // [B, NR, 3]
                                    float* __restrict__ minout,   // [B, NQ]
                                    int NQ, int NR) {
    const int lane  = threadIdx.x & 31;
    const int wid   = blockIdx.x * (blockDim.x >> 5) + (threadIdx.x >> 5);
    const int tilesPerBatch = NQ >> 4;
    const int b     = wid / tilesPerBatch;
    const int q0    = (wid % tilesPerBatch) << 4;
    const int col   = lane & 15;
    const int half  = lane >> 4;   // 0: K=0,1 pair; 1: K=2,3 pair

    // ---- load this lane's query point, build A fragment -------------------
    const float* qrow = Q + ((size_t)b * NQ + q0 + col) * 3;
    const float qx = qrow[0], qy = qrow[1], qz = qrow[2];
    const float q2 = qx * qx + qy * qy + qz * qz;

    v2f afrag;
    afrag.x = half ? (-2.0f * qz) : (-2.0f * qx);
    afrag.y = half ? 1.0f         : (-2.0f * qy);

    // ---- seed accumulator: c0[v] = |q_{v + 8*half}|^2 ---------------------
    // C/D layout: VGPR v, lane l -> row M = v + 8*(l>=16), col N = l&15.
    // Lanes 0..15 hold q2 for points 0..15, so broadcast from lane v+8*half.
    v8f c0;
#pragma unroll
    for (int v = 0; v < 8; ++v)
        c0[v] = __shfl(q2, v + 8 * half, 32);

    // ---- scan all reference tiles -----------------------------------------
    const float* rbase = R + (size_t)b * NR * 3;
    float rmin[8];
#pragma unroll
    for (int v = 0; v < 8; ++v) rmin[v] = 3.402823e38f;

    for (int m0 = 0; m0 < NR; m0 += 16) {
        const float* rrow = rbase + (size_t)(m0 + col) * 3;
        const float rx = rrow[0], ry = rrow[1], rz = rrow[2];
        const float r2 = rx * rx + ry * ry + rz * rz;

        v2f bfrag;
        bfrag.x = half ? rz : rx;
        bfrag.y = half ? r2 : ry;

        v8f c = c0;
        c = __builtin_amdgcn_wmma_f32_16x16x4_f32(
                /*neg_a=*/false, afrag, /*neg_b=*/false, bfrag,
                /*c_mod=*/(short)0, c, /*reuse_a=*/false, /*reuse_b=*/false);

#pragma unroll
        for (int v = 0; v < 8; ++v)
            rmin[v] = fminf(rmin[v], c[v]);
    }

    // ---- min across the 16 lanes of each half (columns), then sqrt --------
#pragma unroll
    for (int v = 0; v < 8; ++v) {
        float r = rmin[v];
        r = fminf(r, __shfl_xor(r, 1, 32));
        r = fminf(r, __shfl_xor(r, 2, 32));
        r = fminf(r, __shfl_xor(r, 4, 32));
        r = fminf(r, __shfl_xor(r, 8, 32));
        rmin[v] = r;
    }
    if (col == 0) {  // lane 0 writes rows 0..7, lane 16 writes rows 8..15
#pragma unroll
        for (int v = 0; v < 8; ++v) {
            const float d = __builtin_sqrtf(fmaxf(rmin[v], 0.0f));
            minout[(size_t)b * NQ + q0 + v + 8 * half] = d;
        }
    }
}

// mean(a[0..na)) + mean(b[0..nb)) -> out[0], single 256-thread block
__global__ void chamfer_mean2(const float* __restrict__ a, int na,
                              const float* __restrict__ b, int nb,
                              float* __restrict__ out) {
    __shared__ float sh[256];
    const float sa = 1.0f / (float)na;
    const float sb = 1.0f / (float)nb;
    float s = 0.0f;
    for (int i = threadIdx.x; i < na; i += 256) s += a[i] * sa;
    for (int i = threadIdx.x; i < nb; i += 256) s += b[i] * sb;
    sh[threadIdx.x] = s;
    __syncthreads();
    for (int off = 128; off > 0; off >>= 1) {
        if ((int)threadIdx.x < off) sh[threadIdx.x] += sh[threadIdx.x + off];
        __syncthreads();
    }
    if (threadIdx.x == 0) out[0] = sh[0];
}

extern "C" void kernel_launch(void* const* d_in, const int* in_sizes, int n_in,
                              void* d_out, int out_size, void* d_ws, size_t ws_size,
                              hipStream_t stream) {
    const float* pred = (const float*)d_in[0];  // [B, N, 3]
    const float* tgt  = (const float*)d_in[1];  // [B, M, 3]

    const int B = 8;
    const int N = in_sizes[0] / (B * 3);
    const int M = in_sizes[1] / (B * 3);

    float* min_p2t = (float*)d_ws;              // B*N floats
    float* min_t2p = min_p2t + (size_t)B * N;   // B*M floats

    const int WAVES_PER_BLOCK = 8;              // 256 threads = 8 wave32
    dim3 block(32 * WAVES_PER_BLOCK);

    // Pass 1: for each pred point, min distance to targets
    {
        const int waves = B * (N / 16);
        chamfer_rowmin_wmma<<<waves / WAVES_PER_BLOCK, block, 0, stream>>>(
            pred, tgt, min_p2t, N, M);
    }
    // Pass 2 (symmetric): for each target point, min distance to preds
    {
        const int waves = B * (M / 16);
        chamfer_rowmin_wmma<<<waves / WAVES_PER_BLOCK, block, 0, stream>>>(
            tgt, pred, min_t2p, M, N);
    }
    // Final scalar: mean(min_p2t) + mean(min_t2p)
    chamfer_mean2<<<1, 256, 0, stream>>>(min_p2t, B * N, min_t2p, B * M,
                                         (float*)d_out);
}